// GATLayer_55061480735372
// MI455X (gfx1250) — compile-verified
//
#include <hip/hip_runtime.h>
#include <math.h>

typedef __attribute__((ext_vector_type(2))) float v2f;
typedef __attribute__((ext_vector_type(8))) float v8f;

namespace {
constexpr int SIZE   = 256;
constexpr int HEADS  = 8;
constexpr int NN     = 500000;
constexpr int NB1    = 1024;                 // blocks, e-pass
constexpr int NB2    = 512;                  // blocks, agg-pass
constexpr int NTILES = NN / 16;              // 31250 16-neighbor tiles
constexpr int NCHUNK = NN / 4;               // 125000 K=4 chunks
constexpr int TOTW1  = NB1 * 8;              // waves in e-pass
constexpr int SPAN1  = (NTILES + TOTW1 - 1) / TOTW1;   // 4
constexpr int TOTP   = NB2 * 4;              // wave-pairs in agg-pass
constexpr int SPAN2  = (NCHUNK + TOTP - 1) / TOTP;     // 62

// workspace layout (float elements)
constexpr size_t WS_AWPAD = 0;                                   // 16 x 256 (heads padded)
constexpr size_t WS_ESELF = WS_AWPAD + 16 * SIZE;                // 16
constexpr size_t WS_E     = WS_ESELF + 16;                       // NN x 8  (e[n][h])
constexpr size_t WS_MAXP  = WS_E + (size_t)NN * HEADS;           // NB1 x 8
constexpr size_t WS_GMAX  = WS_MAXP + (size_t)NB1 * HEADS;       // 16
constexpr size_t WS_AGGP  = WS_GMAX + 16;                        // NB2 x 8 x 256
constexpr size_t WS_SUMP  = WS_AGGP + (size_t)NB2 * HEADS * SIZE;// NB2 x 8
} // namespace

// ---------------- tiny prep: aWpad[16][256], e_self[16] --------------------
__global__ __launch_bounds__(256) void k_prep(const float* __restrict__ x,
                                              const float* __restrict__ W,
                                              const float* __restrict__ a,
                                              float* __restrict__ ws) {
  __shared__ float xs[SIZE];
  __shared__ float red[256];
  const int t = threadIdx.x;
  xs[t] = x[t];
  __syncthreads();

  // aWpad[h][i] = sum_o a_neigh[h][o] * W[h][o][i]; rows 8..15 zero
  for (int h = 0; h < HEADS; ++h) {
    float acc = 0.f;
    const float* an = a + h * (2 * SIZE) + SIZE;       // a_neigh
    const float* Wh = W + (size_t)h * SIZE * SIZE + t; // column t, coalesced across threads
    for (int o = 0; o < SIZE; ++o) acc += an[o] * Wh[(size_t)o * SIZE];
    ws[WS_AWPAD + (size_t)h * SIZE + t] = acc;
  }
  for (int h = HEADS; h < 16; ++h) ws[WS_AWPAD + (size_t)h * SIZE + t] = 0.f;

  // e_self[h] = sum_o a_self[h][o] * dot(W[h][o][:], x)
  const int h = t >> 5;              // 8 heads * 32 threads
  float part = 0.f;
  for (int j = 0; j < 8; ++j) {
    const int o = (t & 31) * 8 + j;
    const float* Wr = W + ((size_t)h * SIZE + o) * SIZE;
    float d = 0.f;
    for (int i = 0; i < SIZE; i += 4)
      d += Wr[i] * xs[i] + Wr[i + 1] * xs[i + 1] + Wr[i + 2] * xs[i + 2] + Wr[i + 3] * xs[i + 3];
    part += a[h * (2 * SIZE) + o] * d;
  }
  red[t] = part;
  __syncthreads();
  if (t < HEADS) {
    float s = 0.f;
    for (int k = 0; k < 32; ++k) s += red[t * 32 + k];
    ws[WS_ESELF + t] = s;
  }
  if (t >= HEADS && t < 16) ws[WS_ESELF + t] = 0.f;
}

// ------------- heavy pass 1: e[n][h] via V_WMMA_F32_16X16X4_F32 ------------
__global__ __launch_bounds__(256) void k_eneigh(const float* __restrict__ nx,
                                                float* __restrict__ ws) {
  __shared__ float aw[16 * 257];     // padded stride: bank-conflict free ds reads
  __shared__ float esf[16];
  __shared__ float redmax[8 * 8];    // [wave][head]
  const int t = threadIdx.x, lane = t & 31, wave = t >> 5;
  const int lanelo = lane & 15, lanehi = lane >> 4;

  for (int q = t; q < 16 * SIZE; q += 256)
    aw[(q >> 8) * 257 + (q & 255)] = ws[WS_AWPAD + q];
  if (t < 16) esf[t] = ws[WS_ESELF + t];
  __syncthreads();

  const float esl = esf[lanelo];     // head = lanelo (0 for padded heads)
  float* __restrict__ e = ws + WS_E;
  float vmax = -3.402823466e38f;     // lanes with lanelo>=8 accumulate junk; never read

  const int gw = blockIdx.x * 8 + wave;
  const int start = gw * SPAN1;
  const int end = (start + SPAN1 < NTILES) ? start + SPAN1 : NTILES;

  for (int tile = start; tile < end; ++tile) {
    const int n0 = tile * 16;
    // A: lane -> row n0+lanelo, K = vgpr + 2*lanehi  (global_load_b64, coalesced stream)
    const float* arow = nx + (size_t)(n0 + lanelo) * SIZE + 2 * lanehi;
    if (tile + 1 < end) __builtin_prefetch(arow + 16 * SIZE, 0, 1);
    v8f c = {0.f, 0.f, 0.f, 0.f, 0.f, 0.f, 0.f, 0.f};
#pragma unroll 4
    for (int kb = 0; kb < SIZE; kb += 4) {
      v2f af = *(const v2f*)(arow + kb);
      v2f bf;                                    // B: aW^T, N = head = lanelo
      bf.x = aw[lanelo * 257 + kb + 2 * lanehi];
      bf.y = aw[lanelo * 257 + kb + 2 * lanehi + 1];
      c = __builtin_amdgcn_wmma_f32_16x16x4_f32(false, af, false, bf, (short)0, c,
                                                false, false);
    }
    // D[m,n]: neighbor = n0 + r + 8*lanehi, head = lanelo
    float vals[8];
#pragma unroll
    for (int r = 0; r < 8; ++r) {
      vals[r] = c[r] + esl;
      vmax = fmaxf(vmax, vals[r]);   // unconditional; invalid-head lanes unused later
    }
    if (lanelo < HEADS) {            // single predicated region: 8 plain stores
      float* eb = e + (size_t)(n0 + 8 * lanehi) * HEADS + lanelo;
#pragma unroll
      for (int r = 0; r < 8; ++r) eb[r * HEADS] = vals[r];
    }
  }
  vmax = fmaxf(vmax, __shfl_xor(vmax, 16, 32));
  if (lane < HEADS) redmax[wave * 8 + lane] = vmax;
  __syncthreads();
  if (t < HEADS) {
    float m = redmax[t];
    for (int w2 = 1; w2 < 8; ++w2) m = fmaxf(m, redmax[w2 * 8 + t]);
    ws[WS_MAXP + (size_t)blockIdx.x * HEADS + t] = m;
  }
}

// ---------------- reduce block maxima -> gmax[8] ---------------------------
__global__ void k_redmax(float* __restrict__ ws) {
  const int t = threadIdx.x;
  if (t < HEADS) {
    float m = -3.402823466e38f;
    for (int b = 0; b < NB1; ++b) m = fmaxf(m, ws[WS_MAXP + (size_t)b * HEADS + t]);
    ws[WS_GMAX + t] = m;
  }
  if (t >= HEADS && t < 16) ws[WS_GMAX + t] = 0.f;
}

// ------- heavy pass 2: agg partials + sumexp partials via WMMA -------------
__global__ __launch_bounds__(256) void k_agg(const float* __restrict__ nx,
                                             float* __restrict__ ws) {
  __shared__ float lagg[8][1024];    // per wave: 8 heads x 128 local features (32 KB)
  __shared__ float lsum[8][8];
  const int t = threadIdx.x, lane = t & 31, wave = t >> 5;
  const int lanelo = lane & 15, lanehi = lane >> 4;
  const int h = lanelo;
  const int hc = h & 7;              // clamped head: keeps loads in-bounds, branchless
  const bool hvalid = (h < HEADS);
  const float gm = ws[WS_GMAX + hc];
  const float* __restrict__ e = ws + WS_E;

  const int gw = blockIdx.x * 8 + wave;
  const int pairIdx = gw >> 1;
  const int fhalf = gw & 1;          // which 128-feature half this wave owns
  const int ibase0 = fhalf * 128;
  const int start = pairIdx * SPAN2;
  const int end = (start + SPAN2 < NCHUNK) ? start + SPAN2 : NCHUNK;

  const v8f cz = {0.f, 0.f, 0.f, 0.f, 0.f, 0.f, 0.f, 0.f};
  v8f c[8];
#pragma unroll
  for (int i = 0; i < 8; ++i) c[i] = cz;
  float sacc = 0.f;

  for (int cb = start; cb < end; ++cb) {
    const int nb = cb * 4;
    // A: softmax weights, lane -> head = lanelo, neighbor K = vgpr + 2*lanehi.
    // Branchless: clamped-address loads + cndmask zeroing for padded heads.
    const int n0 = nb + 2 * lanehi;
    const float e0 = e[(size_t)n0 * HEADS + hc];
    const float e1 = e[(size_t)(n0 + 1) * HEADS + hc];
    const float w0 = hvalid ? __expf(e0 - gm) : 0.f;
    const float w1 = hvalid ? __expf(e1 - gm) : 0.f;
    v2f a;
    a.x = w0;
    a.y = w1;
    sacc += w0 + w1;
    // B: raw n_x rows; lanes 0-15 read 64 contiguous bytes (fully coalesced)
    const float* brow0 = nx + (size_t)(nb + 2 * lanehi) * SIZE + lanelo;
    const float* brow1 = brow0 + SIZE;
    if (cb + 1 < end) __builtin_prefetch(brow0 + 4 * SIZE, 0, 1);
#pragma unroll
    for (int tt = 0; tt < 8; ++tt) {
      const int ib = ibase0 + tt * 16;
      v2f b;
      b.x = brow0[ib];
      b.y = brow1[ib];
      c[tt] = __builtin_amdgcn_wmma_f32_16x16x4_f32(false, a, false, b, (short)0,
                                                    c[tt], false, false);
    }
  }

  // sumexp: lanes l and l^16 hold the same head; count each pair's chunks once
  sacc += __shfl_xor(sacc, 16, 32);
  if (lane < HEADS) lsum[wave][lane] = (fhalf == 0) ? sacc : 0.f;

  // C[m,n]: head = r + 8*lanehi (valid rows live in lanes 0-15), feature N = lanelo
  if (lanehi == 0) {
#pragma unroll
    for (int tt = 0; tt < 8; ++tt)
#pragma unroll
      for (int r = 0; r < 8; ++r) lagg[wave][r * 128 + tt * 16 + lanelo] = c[tt][r];
  }
  __syncthreads();

  // deterministic in-block reduction -> per-block partials
  float* __restrict__ aggp = ws + WS_AGGP + (size_t)blockIdx.x * HEADS * SIZE;
  for (int q = t; q < HEADS * SIZE; q += 256) {
    const int hh = q >> 8, ff = q & 255;
    const int fh = ff >> 7, fl = ff & 127;
    float s = 0.f;
    for (int w2 = fh; w2 < 8; w2 += 2) s += lagg[w2][hh * 128 + fl];
    aggp[q] = s;
  }
  if (t < HEADS) {
    float s = 0.f;
    for (int w2 = 0; w2 < 8; ++w2) s += lsum[w2][t];
    ws[WS_SUMP + (size_t)blockIdx.x * HEADS + t] = s;
  }
}

// -------- final: reduce partials, normalize, h = W·agg, sigmoid ------------
__global__ __launch_bounds__(256) void k_final(const float* __restrict__ W,
                                               float* __restrict__ out,
                                               float* __restrict__ ws) {
  __shared__ float aggn[HEADS * SIZE];
  __shared__ float gsum[HEADS];
  const int t = threadIdx.x;
  if (t < HEADS) {
    float s = 0.f;
    for (int b = 0; b < NB2; ++b) s += ws[WS_SUMP + (size_t)b * HEADS + t];
    gsum[t] = s;
  }
  __syncthreads();
  for (int q = t; q < HEADS * SIZE; q += 256) {
    float s = 0.f;
    for (int b = 0; b < NB2; ++b) s += ws[WS_AGGP + (size_t)b * HEADS * SIZE + q];
    aggn[q] = s / gsum[q >> 8];
  }
  __syncthreads();
  for (int h = 0; h < HEADS; ++h) {
    const float* Wr = W + ((size_t)h * SIZE + t) * SIZE;
    const float* ag = aggn + h * SIZE;
    float acc = 0.f;
    for (int i = 0; i < SIZE; i += 4)
      acc += Wr[i] * ag[i] + Wr[i + 1] * ag[i + 1] + Wr[i + 2] * ag[i + 2] +
             Wr[i + 3] * ag[i + 3];
    out[(size_t)t * HEADS + h] = 1.f / (1.f + __expf(-acc)); // out.T layout (256,8)
  }
}

extern "C" void kernel_launch(void* const* d_in, const int* in_sizes, int n_in,
                              void* d_out, int out_size, void* d_ws, size_t ws_size,
                              hipStream_t stream) {
  (void)in_sizes; (void)n_in; (void)out_size; (void)ws_size;
  const float* x  = (const float*)d_in[0];
  const float* nx = (const float*)d_in[1];
  const float* W  = (const float*)d_in[2];
  const float* a  = (const float*)d_in[3];
  float* ws  = (float*)d_ws;
  float* out = (float*)d_out;

  hipLaunchKernelGGL(k_prep,   dim3(1),   dim3(256), 0, stream, x, W, a, ws);
  hipLaunchKernelGGL(k_eneigh, dim3(NB1), dim3(256), 0, stream, nx, ws);
  hipLaunchKernelGGL(k_redmax, dim3(1),   dim3(64),  0, stream, ws);
  hipLaunchKernelGGL(k_agg,    dim3(NB2), dim3(256), 0, stream, nx, ws);
  hipLaunchKernelGGL(k_final,  dim3(1),   dim3(256), 0, stream, W, out, ws);
}